// GCNNFingerprintRecognizer_77146202571273
// MI455X (gfx1250) — compile-verified
//
#include <hip/hip_runtime.h>
#include <hip/hip_bf16.h>

#define N_NODES 100000
#define N_EDGES 1600000
#define NTILES  (N_NODES / 16)   // 6250, exact

typedef __attribute__((ext_vector_type(2))) float v2f;
typedef __attribute__((ext_vector_type(8))) float v8f;

// ---------------------------------------------------------------------------
// Zero fill (workspace accumulators must be zeroed every call)
// ---------------------------------------------------------------------------
__global__ void zero_f4_kernel(float4* __restrict__ p, int n4) {
    int i = blockIdx.x * blockDim.x + threadIdx.x;
    if (i < n4) p[i] = float4{0.f, 0.f, 0.f, 0.f};
}

// ---------------------------------------------------------------------------
// Edge scatter-add: agg[dst[e]][:] += feat[src[e]][:]
// One thread per (edge, 4-float chunk); float4 gather + 4 f32 atomics.
// ---------------------------------------------------------------------------
template <int DIM>
__global__ void scatter_add_kernel(const float* __restrict__ feat,
                                   const int* __restrict__ src,
                                   const int* __restrict__ dst,
                                   float* __restrict__ agg) {
    constexpr int NC = DIM / 4;
    int tid = blockIdx.x * blockDim.x + threadIdx.x;
    int e = tid / NC;
    int c = tid % NC;
    if (e >= N_EDGES) return;
    int s = src[e];
    int d = dst[e];
    const float4 v = *(const float4*)(feat + (long long)s * DIM + c * 4);
    float* o = agg + (long long)d * DIM + c * 4;
    atomicAdd(o + 0, v.x);
    atomicAdd(o + 1, v.y);
    atomicAdd(o + 2, v.z);
    atomicAdd(o + 3, v.w);
}

// ---------------------------------------------------------------------------
// WMMA helpers (f32 16x16x4).
// A (16x4): lane l<16 holds row M=l, K = {k0, k0+1}; lane l>=16 holds row
//           M=l-16, K = {k0+2, k0+3}  (ISA 7.12.2, 32-bit A 16x4).
// B (4x16): VGPR0 holds rows K=k0 (lanes 0-15) / K=k0+2 (lanes 16-31);
//           VGPR1 holds K=k0+1 / K=k0+3 (row-striped like C/D).
// C/D (16x16): VGPR r = row M=r (lanes 0-15) / M=r+8 (lanes 16-31).
// ---------------------------------------------------------------------------
__device__ __forceinline__ v8f wmma_f32_k4(v2f a, v2f b, v8f c) {
    return __builtin_amdgcn_wmma_f32_16x16x4_f32(
        false, a, false, b, (short)0, c, false, false);
}

// ---------------------------------------------------------------------------
// Layer 1: h1[16-row tile] = agg1 @ W_rel1 + x @ W_root1 + b1    ([N,16]->[N,32])
// One wave per tile; 2 col-tiles; K=16 -> 4 WMMA per operand per col-tile.
// ---------------------------------------------------------------------------
__global__ __launch_bounds__(128) void gemm1_kernel(
    const float* __restrict__ agg1, const float* __restrict__ x,
    const float* __restrict__ Wrel, const float* __restrict__ Wroot,
    const float* __restrict__ b1, float* __restrict__ h1) {
    const int wave = threadIdx.x >> 5;
    const int lane = threadIdx.x & 31;
    const int tile = blockIdx.x * 4 + wave;
    if (tile >= NTILES) return;               // wave-uniform exit
    const int row0 = tile * 16;
    const int half = lane >> 4;
    const int lr   = lane & 15;

#pragma unroll
    for (int ct = 0; ct < 2; ++ct) {
        const int col = ct * 16 + lr;
        v8f c = {};
#pragma unroll
        for (int k0 = 0; k0 < 16; k0 += 4) {
            const int kb = k0 + 2 * half;
            v2f a = *(const v2f*)(agg1 + (row0 + lr) * 16 + kb);
            v2f b;
            b.x = Wrel[kb * 32 + col];
            b.y = Wrel[(kb + 1) * 32 + col];
            c = wmma_f32_k4(a, b, c);
        }
#pragma unroll
        for (int k0 = 0; k0 < 16; k0 += 4) {
            const int kb = k0 + 2 * half;
            v2f a = *(const v2f*)(x + (row0 + lr) * 16 + kb);
            v2f b;
            b.x = Wroot[kb * 32 + col];
            b.y = Wroot[(kb + 1) * 32 + col];
            c = wmma_f32_k4(a, b, c);
        }
        const float bias = b1[col];
#pragma unroll
        for (int r = 0; r < 8; ++r)
            h1[(row0 + r + 8 * half) * 32 + col] = c[r] + bias;
    }
}

// ---------------------------------------------------------------------------
// Layer 2 + FC (fused): h2 = agg2 @ W_rel2 + h1 @ W_root2 + b2   ([N,64])
//                       out = h2 @ Wfc(pad 64x16) + b_fc          ([N,10])
// h2 tile staged through LDS to convert D-layout -> A-layout for 2nd GEMM.
// ---------------------------------------------------------------------------
__global__ __launch_bounds__(128) void gemm2_fc_kernel(
    const float* __restrict__ agg2, const float* __restrict__ h1,
    const float* __restrict__ Wrel, const float* __restrict__ Wroot,
    const float* __restrict__ b2, const float* __restrict__ Wfc,
    const float* __restrict__ bfc, float* __restrict__ out) {
    __shared__ float lds[4][16 * 64];
    const int wave = threadIdx.x >> 5;
    const int lane = threadIdx.x & 31;
    const int tile = blockIdx.x * 4 + wave;
    if (tile >= NTILES) return;               // wave-uniform exit
    const int row0 = tile * 16;
    const int half = lane >> 4;
    const int lr   = lane & 15;
    float* tl = lds[wave];

    // ---- h2 = agg2 @ Wrel2 + h1 @ Wroot2 + b2 (four 16x16 col-tiles) ----
#pragma unroll
    for (int ct = 0; ct < 4; ++ct) {
        const int col = ct * 16 + lr;
        v8f c = {};
#pragma unroll
        for (int k0 = 0; k0 < 32; k0 += 4) {
            const int kb = k0 + 2 * half;
            v2f a = *(const v2f*)(agg2 + (row0 + lr) * 32 + kb);
            v2f b;
            b.x = Wrel[kb * 64 + col];
            b.y = Wrel[(kb + 1) * 64 + col];
            c = wmma_f32_k4(a, b, c);
        }
#pragma unroll
        for (int k0 = 0; k0 < 32; k0 += 4) {
            const int kb = k0 + 2 * half;
            v2f a = *(const v2f*)(h1 + (row0 + lr) * 32 + kb);
            v2f b;
            b.x = Wroot[kb * 64 + col];
            b.y = Wroot[(kb + 1) * 64 + col];
            c = wmma_f32_k4(a, b, c);
        }
        const float bias = b2[col];
#pragma unroll
        for (int r = 0; r < 8; ++r)
            tl[(r + 8 * half) * 64 + col] = c[r] + bias;
    }
    // same-wave LDS produce -> consume: drain DS counter
    asm volatile("s_wait_dscnt 0" ::: "memory");

    // ---- out = h2 @ Wfc (zero-padded 64x16) + bfc; K = 64 ----
    // Mask instead of branch so EXEC stays all-ones across every WMMA.
    const float fcmask = (lr < 10) ? 1.0f : 0.0f;
    const int   fccol  = (lr < 10) ? lr : 9;
    v8f c = {};
#pragma unroll
    for (int k0 = 0; k0 < 64; k0 += 4) {
        const int kb = k0 + 2 * half;
        v2f a = *(const v2f*)(tl + lr * 64 + kb);
        v2f b;
        b.x = Wfc[kb * 10 + fccol] * fcmask;
        b.y = Wfc[(kb + 1) * 10 + fccol] * fcmask;
        c = wmma_f32_k4(a, b, c);
    }
    if (lr < 10) {
        const float bias = bfc[lr];
#pragma unroll
        for (int r = 0; r < 8; ++r)
            out[(row0 + r + 8 * half) * 10 + lr] = c[r] + bias;
    }
}

// ---------------------------------------------------------------------------
// Host launcher
// ---------------------------------------------------------------------------
extern "C" void kernel_launch(void* const* d_in, const int* in_sizes, int n_in,
                              void* d_out, int out_size, void* d_ws, size_t ws_size,
                              hipStream_t stream) {
    const float* x     = (const float*)d_in[0];
    const int*   ei    = (const int*)d_in[1];
    const float* Wrel1 = (const float*)d_in[2];
    const float* Wroot1= (const float*)d_in[3];
    const float* b1    = (const float*)d_in[4];
    const float* Wrel2 = (const float*)d_in[5];
    const float* Wroot2= (const float*)d_in[6];
    const float* b2    = (const float*)d_in[7];
    const float* Wfc   = (const float*)d_in[8];
    const float* bfc   = (const float*)d_in[9];
    float* outp = (float*)d_out;

    const int* src = ei;                 // edge_index[0]
    const int* dst = ei + N_EDGES;       // edge_index[1]

    // workspace layout: agg1 [N,16] | h1 [N,32] | agg2 [N,32]  (32 MB total)
    float* ws   = (float*)d_ws;
    float* agg1 = ws;
    float* h1   = ws + (size_t)N_NODES * 16;
    float* agg2 = ws + (size_t)N_NODES * 48;

    // zero accumulators (h1 fully overwritten, no need to zero it)
    {
        int n4 = N_NODES * 16 / 4;
        zero_f4_kernel<<<(n4 + 255) / 256, 256, 0, stream>>>((float4*)agg1, n4);
        n4 = N_NODES * 32 / 4;
        zero_f4_kernel<<<(n4 + 255) / 256, 256, 0, stream>>>((float4*)agg2, n4);
    }

    // layer 1 aggregation + GEMM
    {
        int nthreads = N_EDGES * 4;      // 4 chunks of float4 per edge (dim 16)
        scatter_add_kernel<16><<<(nthreads + 255) / 256, 256, 0, stream>>>(x, src, dst, agg1);
        gemm1_kernel<<<(NTILES + 3) / 4, 128, 0, stream>>>(agg1, x, Wrel1, Wroot1, b1, h1);
    }

    // layer 2 aggregation + fused GEMM + FC head
    {
        int nthreads = N_EDGES * 8;      // 8 chunks per edge (dim 32)
        scatter_add_kernel<32><<<(nthreads + 255) / 256, 256, 0, stream>>>(h1, src, dst, agg2);
        gemm2_fc_kernel<<<(NTILES + 3) / 4, 128, 0, stream>>>(agg2, h1, Wrel2, Wroot2, b2,
                                                              Wfc, bfc, outp);
    }
}